// SpecAugmentLB_3985729650733
// MI455X (gfx1250) — compile-verified
//
#include <hip/hip_runtime.h>

// SpecAugment: time-warp (shifted gather, zero borders) + freq mask + time mask.
// spec: [B=128, M=80, T=3000] fp32, T contiguous.
// out[b,m,t] = fkeep(m)*tkeep(t)*valid(b,t) * spec[b,m, t<c_b ? t-W : t+W]

#define SA_B 128
#define SA_M 80
#define SA_T 3000
#define SA_W 80
#define SA_T4 (SA_T / 4)   // 750, exact

typedef __attribute__((ext_vector_type(4))) float v4f;  // native vector for b128 + NT store

__global__ __launch_bounds__(256) void specaugment_kernel(
    const float* __restrict__ spec,
    const int*   __restrict__ centers,
    const int*   __restrict__ f_start_p,
    const int*   __restrict__ f_width_p,
    const int*   __restrict__ t_start_p,
    const int*   __restrict__ t_width_p,
    float*       __restrict__ out) {

    const int bm = blockIdx.y;              // b*M + m, 0..10239
    const int b  = bm / SA_M;
    const int m  = bm - b * SA_M;
    const int tv = blockIdx.x * blockDim.x + threadIdx.x;  // float4 group id
    if (tv >= SA_T4) return;
    const int t0 = tv * 4;

    float* dst = out + (size_t)bm * SA_T + t0;

    // Uniform scalar parameters (1-element device arrays).
    const int fs = *f_start_p;
    const int fw = *f_width_p;
    const int ts = *t_start_p;
    const int tw = *t_width_p;

    v4f v;

    // Frequency mask: whole row is zero -> skip all loads.
    if (m >= fs && m < fs + fw) {
        v = (v4f)(0.f);
        __builtin_nontemporal_store(v, (v4f*)dst);
        return;
    }

    const int c = centers[b];               // in [W, T-W]
    const float* row = spec + (size_t)bm * SA_T;

    if (t0 + 3 < c) {
        // Entire group below center: src = t - W (src is a multiple of 4).
        const int src = t0 - SA_W;
        if (src >= 0) {
            v = *(const v4f*)(row + src);
        } else {
            // src <= -4 -> all four lanes out of range -> zero.
            v = (v4f)(0.f);
        }
    } else if (t0 >= c) {
        // Entire group at/above center: src = t + W (multiple of 4).
        const int src = t0 + SA_W;
        if (src + 3 < SA_T) {
            v = *(const v4f*)(row + src);
        } else {
            // src (multiple of 4) >= T -> all four invalid -> zero.
            v = (v4f)(0.f);
        }
    } else {
        // Group straddles the warp center (at most one group per row).
#pragma unroll
        for (int k = 0; k < 4; ++k) {
            const int t   = t0 + k;
            const int idx = (t < c) ? (t - SA_W) : (t + SA_W);
            v[k] = (idx >= 0 && idx < SA_T) ? row[idx] : 0.f;
        }
    }

    // Time mask: zero t in [ts, ts+tw).
    const int te = ts + tw;
    if (t0 >= ts && t0 + 3 < te) {
        v = (v4f)(0.f);
    } else if (t0 + 3 >= ts && t0 < te) {
#pragma unroll
        for (int k = 0; k < 4; ++k) {
            const int t = t0 + k;
            if (t >= ts && t < te) v[k] = 0.f;
        }
    }

    __builtin_nontemporal_store(v, (v4f*)dst);
}

extern "C" void kernel_launch(void* const* d_in, const int* in_sizes, int n_in,
                              void* d_out, int out_size, void* d_ws, size_t ws_size,
                              hipStream_t stream) {
    const float* spec    = (const float*)d_in[0];
    const int*   centers = (const int*)d_in[1];
    const int*   f_start = (const int*)d_in[2];
    const int*   f_width = (const int*)d_in[3];
    const int*   t_start = (const int*)d_in[4];
    const int*   t_width = (const int*)d_in[5];
    float*       out     = (float*)d_out;

    dim3 block(256);                              // 8 wave32 waves
    dim3 grid((SA_T4 + 255) / 256, SA_B * SA_M);  // (3, 10240)
    specaugment_kernel<<<grid, block, 0, stream>>>(
        spec, centers, f_start, f_width, t_start, t_width, out);
}